// ScoreClassify_29970281791567
// MI455X (gfx1250) — compile-verified
//
#include <hip/hip_runtime.h>
#include <math.h>

#define BB 4096
#define KK 512
#define TT 30
#define NTHREADS 512
#define NWAVES (NTHREADS / 32)

typedef __attribute__((ext_vector_type(2))) float v2f;
typedef __attribute__((ext_vector_type(8))) float v8f;

__device__ __forceinline__ float waveSum(float v) {
#pragma unroll
    for (int o = 16; o > 0; o >>= 1) v += __shfl_xor(v, o, 32);
    return v;
}
__device__ __forceinline__ float waveMax(float v) {
#pragma unroll
    for (int o = 16; o > 0; o >>= 1) v = fmaxf(v, __shfl_xor(v, o, 32));
    return v;
}

// Pass 1: one block per batch row b. Computes cls_loss[b] + miss_loss[b] -> partial[b].
__global__ __launch_bounds__(NTHREADS)
void score_loss_pass1(const float* __restrict__ output,
                      const float* __restrict__ cand,
                      const float* __restrict__ gt,
                      const float* __restrict__ scales,
                      float* __restrict__ partial) {
    __shared__ float s_r0[NWAVES];
    __shared__ float s_r1[NWAVES];
    __shared__ float s_r2[NWAVES];
    __shared__ int   s_ri[NWAVES];
    __shared__ float s_out[KK];
    __shared__ float s_fde[KK];
    __shared__ float s_bcast[3];

    const int b    = blockIdx.x;
    const int k    = threadIdx.x;
    const int lane = k & 31;
    const int wid  = k >> 5;

    // ---- gather: only the final timestep matters ----
    const float o = output[(size_t)b * KK + k];
    const size_t cidx = (((size_t)b * KK + k) * TT + (TT - 1)) * 2;  // 8B aligned
    const size_t gidx = ((size_t)b * TT + (TT - 1)) * 2;
    const float dx = cand[cidx]     - gt[gidx];
    const float dy = cand[cidx + 1] - gt[gidx + 1];
    const float fde   = scales[b] * sqrtf(dx * dx + dy * dy);
    const float score = -fde;  // TEMP == 1.0

    s_out[k] = o;
    s_fde[k] = fde;

    // ---- block max of score (for target softmax) and out (for log_softmax) ----
    float m1 = waveMax(score);
    float m2 = waveMax(o);
    if (lane == 0) { s_r0[wid] = m1; s_r1[wid] = m2; }
    __syncthreads();
    if (wid == 0) {
        float x1 = (lane < NWAVES) ? s_r0[lane] : -INFINITY;
        float x2 = (lane < NWAVES) ? s_r1[lane] : -INFINITY;
        x1 = waveMax(x1);
        x2 = waveMax(x2);
        if (lane == 0) { s_bcast[0] = x1; s_bcast[1] = x2; }
    }
    __syncthreads();
    m1 = s_bcast[0];
    m2 = s_bcast[1];

    // ---- three sums: S = sum e, A = sum e*out, Z = sum exp(out - m2) ----
    const float e  = expf(score - m1);
    const float eo = e * o;
    const float z  = expf(o - m2);
    float S = waveSum(e), A = waveSum(eo), Z = waveSum(z);
    if (lane == 0) { s_r0[wid] = S; s_r1[wid] = A; s_r2[wid] = Z; }
    __syncthreads();
    if (wid == 0) {
        float x1 = (lane < NWAVES) ? s_r0[lane] : 0.0f;
        float x2 = (lane < NWAVES) ? s_r1[lane] : 0.0f;
        float x3 = (lane < NWAVES) ? s_r2[lane] : 0.0f;
        x1 = waveSum(x1);
        x2 = waveSum(x2);
        x3 = waveSum(x3);
        if (lane == 0) { s_bcast[0] = x1; s_bcast[1] = x2; s_bcast[2] = x3; }
    }
    __syncthreads();
    S = s_bcast[0];
    A = s_bcast[1];
    Z = s_bcast[2];

    const float lse = m2 + logf(Z);
    const float cls = lse - A / S;  // = -(tgt * log_softmax(out)).sum()

    // ---- top-6 of out, min fde among them (tie-break: lower index, as lax.top_k) ----
    float minFde = INFINITY;  // only thread 0's copy is meaningful
    for (int g = 0; g < 6; ++g) {
        float v = s_out[k];
        int   i = k;
#pragma unroll
        for (int off = 16; off > 0; off >>= 1) {
            float ov = __shfl_xor(v, off, 32);
            int   oi = __shfl_xor(i, off, 32);
            if (ov > v || (ov == v && oi < i)) { v = ov; i = oi; }
        }
        if (lane == 0) { s_r0[wid] = v; s_ri[wid] = i; }
        __syncthreads();
        if (k == 0) {
            float bv = s_r0[0];
            int   bi = s_ri[0];
            for (int w = 1; w < NWAVES; ++w) {
                float wv = s_r0[w];
                int   wi = s_ri[w];
                if (wv > bv || (wv == bv && wi < bi)) { bv = wv; bi = wi; }
            }
            minFde = fminf(minFde, s_fde[bi]);
            s_out[bi] = -INFINITY;  // exclude for next round
        }
        __syncthreads();
    }

    if (k == 0) {
        const float miss = fmaxf(minFde - 2.0f, 0.0f);
        partial[b] = cls + miss;
    }
}

// Pass 2: single wave32. Deterministic mean over 4096 partials, summed through the
// matrix pipe: D = A(16x4 chunk) * ones(4x16) + C accumulates exact f32 row sums.
__global__ __launch_bounds__(32)
void score_loss_pass2(const float* __restrict__ partial, float* __restrict__ out) {
    const int lane = threadIdx.x;  // 0..31, full wave -> EXEC all ones for WMMA
    v8f c = {};
    v2f ones;
    ones.x = 1.0f;
    ones.y = 1.0f;

    // 4096 / 64 = 64 chunks; lane l contributes A elements {2l, 2l+1} of each chunk.
    for (int chunk = 0; chunk < BB / 64; ++chunk) {
        const float* p = partial + chunk * 64 + lane * 2;
        v2f a;
        a.x = p[0];
        a.y = p[1];
        // D[m,n] = sum_k A[m,k] + C[m,n]  (B == ones)
        c = __builtin_amdgcn_wmma_f32_16x16x4_f32(
            /*neg_a=*/false, a, /*neg_b=*/false, ones,
            /*c_mod=*/(short)0, c, /*reuse_a=*/false, /*reuse_b=*/false);
    }

    // Column N=0 lives in lane 0 (rows 0..7 across the 8 C VGPRs) and lane 16 (rows 8..15).
    float t = 0.0f;
#pragma unroll
    for (int j = 0; j < 8; ++j) {
        float cj = c[j];
        t += __shfl(cj, 0, 32) + __shfl(cj, 16, 32);
    }
    if (lane == 0) out[0] = t / (float)BB;
}

extern "C" void kernel_launch(void* const* d_in, const int* in_sizes, int n_in,
                              void* d_out, int out_size, void* d_ws, size_t ws_size,
                              hipStream_t stream) {
    const float* output = (const float*)d_in[0];   // (B*K, 1)
    const float* cand   = (const float*)d_in[1];   // (B, K, T, 2)
    const float* gt     = (const float*)d_in[2];   // (B, T, 2)
    const float* scales = (const float*)d_in[3];   // (B,)
    float* out = (float*)d_out;                    // scalar
    float* partial = (float*)d_ws;                 // B floats of scratch

    score_loss_pass1<<<BB, NTHREADS, 0, stream>>>(output, cand, gt, scales, partial);
    score_loss_pass2<<<1, 32, 0, stream>>>(partial, out);
}